// HCHA_23493471109502
// MI455X (gfx1250) — compile-verified
//
#include <hip/hip_runtime.h>

typedef __attribute__((ext_vector_type(2))) float v2f;
typedef __attribute__((ext_vector_type(8))) float v8f;

// ---------------- helpers ----------------
__device__ __forceinline__ unsigned f2mono(float f) {
  unsigned u = __float_as_uint(f);
  return (u & 0x80000000u) ? ~u : (u | 0x80000000u);
}
__device__ __forceinline__ float mono2f(unsigned u) {
  return (u & 0x80000000u) ? __uint_as_float(u & 0x7fffffffu) : __uint_as_float(~u);
}

__global__ void fill_f32(float* __restrict__ p, long n, float v) {
  long i = (long)blockIdx.x * blockDim.x + threadIdx.x;
  long st = (long)gridDim.x * blockDim.x;
  for (; i < n; i += st) p[i] = v;
}

// ---------------- fp32 WMMA GEMM: C[M,NCOLS] = A[M,K] * W[K,NCOLS] ----------------
// Requires M % 16 == 0 (true for all shapes here: 50000, 10000).
// One block computes a 16-row strip. A staged in LDS as [K][16] (transposed):
// wmma A reads are 16 consecutive lanes over 16 consecutive floats -> bank-conflict-free.
// Per-wave 16x16 tile, K advanced 4/step with V_WMMA_F32_16X16X4_F32.
// A 16x4 layout (ISA 7.12.2): lanes 0-15 -> {K0,K1} in v[0],v[1]; lanes 16-31 -> {K2,K3}.
// C/D layout: VGPR r: lanes0-15 -> M=r, N=lane; lanes16-31 -> M=r+8, N=lane-16.
template <int K, int NCOLS>
__global__ __launch_bounds__(256) void wmma_gemm(const float* __restrict__ A,
                                                 const float* __restrict__ W,
                                                 float* __restrict__ C, int M) {
  constexpr int NT = NCOLS / 16;
  __shared__ float Alds[K][16];
  const int m0 = blockIdx.x * 16;
  const int tid = threadIdx.x;
  for (int i = tid; i < 16 * K; i += blockDim.x) {
    int r = i / K, k = i % K;               // coalesced on k
    Alds[k][r] = A[(size_t)(m0 + r) * K + k];
  }
  __syncthreads();
  const int wave = tid >> 5;
  const int lane = tid & 31;
  const int l = lane & 15;
  const bool hi = lane >= 16;
  const int nwaves = blockDim.x >> 5;
  for (int nt = wave; nt < NT; nt += nwaves) {
    const int n0 = nt * 16;
    v8f acc = {0.f, 0.f, 0.f, 0.f, 0.f, 0.f, 0.f, 0.f};
#pragma unroll 4
    for (int k0 = 0; k0 < K; k0 += 4) {
      const int ka = k0 + (hi ? 2 : 0);
      v2f a, b;
      a.x = Alds[ka][l];
      a.y = Alds[ka + 1][l];
      const float* wp = W + (size_t)ka * NCOLS + n0 + l;  // B 4x16 mirrors A layout
      b.x = wp[0];
      b.y = wp[NCOLS];
      acc = __builtin_amdgcn_wmma_f32_16x16x4_f32(false, a, false, b, (short)0, acc,
                                                  false, false);
    }
    // unguarded store: single base pointer + constant row offsets -> store clause
    float* cp = C + (size_t)(m0 + (hi ? 8 : 0)) * NCOLS + n0 + l;
#pragma unroll
    for (int r = 0; r < 8; ++r) cp[(size_t)r * NCOLS] = acc[r];
  }
}

// ---------------- attention dot: out[row,h] = sum_f feat[row, h*64+f] * att[h, att_off+f] ----
__global__ void att_dot(const float* __restrict__ feat, const float* __restrict__ att,
                        float* __restrict__ out, int rows, int att_off) {
  int row = blockIdx.x * (blockDim.x >> 5) + (threadIdx.x >> 5);
  if (row >= rows) return;
  int lane = threadIdx.x & 31;
  int c = lane * 8;            // 8 contiguous channels per lane, head = lane/8
  int head = c >> 6;
  const float* fp = feat + (size_t)row * 256 + c;
  const float* ap = att + head * 128 + att_off + (c & 63);
  float p = 0.f;
#pragma unroll
  for (int i = 0; i < 8; ++i) p += fp[i] * ap[i];
  p += __shfl_xor(p, 1, 8);
  p += __shfl_xor(p, 2, 8);
  p += __shfl_xor(p, 4, 8);
  if ((lane & 7) == 0) out[row * 4 + head] = p;
}

// ---------------- degree counting ----------------
__global__ void degree_count(const int* __restrict__ node, const int* __restrict__ hedge,
                             float* __restrict__ Dc, float* __restrict__ Bc, int nnz) {
  int k = blockIdx.x * blockDim.x + threadIdx.x;
  if (k >= nnz) return;
  atomicAdd(&Dc[node[k]], 1.f);
  atomicAdd(&Bc[hedge[k]], 1.f);
}

// ---------------- softmax phase 1: leaky-relu logits + segment max ----------------
__global__ void logit_max(const int* __restrict__ node, const int* __restrict__ hedge,
                          const float* __restrict__ an, const float* __restrict__ ae,
                          float* __restrict__ logits, unsigned* __restrict__ mu, int nnz) {
  int k = blockIdx.x * blockDim.x + threadIdx.x;
  if (k >= nnz) return;
  int nd = node[k], he = hedge[k];
#pragma unroll
  for (int h = 0; h < 4; ++h) {
    float v = an[nd * 4 + h] + ae[he * 4 + h];
    v = (v > 0.f) ? v : 0.2f * v;            // leaky_relu, slope 0.2
    logits[k * 4 + h] = v;
    atomicMax(&mu[he * 4 + h], f2mono(v));
  }
}

// ---------------- softmax phase 2: exp + segment sum ----------------
__global__ void exp_sum(const int* __restrict__ hedge, float* __restrict__ logits,
                        const unsigned* __restrict__ mu, float* __restrict__ s, int nnz) {
  int k = blockIdx.x * blockDim.x + threadIdx.x;
  if (k >= nnz) return;
  int he = hedge[k];
#pragma unroll
  for (int h = 0; h < 4; ++h) {
    unsigned u = mu[he * 4 + h];
    float m = u ? mono2f(u) : 0.f;           // empty segment -> 0 (matches isfinite guard)
    float ex = __expf(logits[k * 4 + h] - m);
    logits[k * 4 + h] = ex;
    atomicAdd(&s[he * 4 + h], ex);
  }
}

__global__ void inv_pos(float* __restrict__ p, long n) {
  long i = (long)blockIdx.x * blockDim.x + threadIdx.x;
  long st = (long)gridDim.x * blockDim.x;
  for (; i < n; i += st) { float v = p[i]; p[i] = v > 0.f ? 1.f / v : 0.f; }
}

// ---------------- layer 1 propagate: nodes -> hyperedges (256 ch, wave/nnz) -------
__global__ void prop1_l1(const int* __restrict__ node, const int* __restrict__ hedge,
                         const float* __restrict__ xh, float* __restrict__ alpha,
                         const float* __restrict__ s, const float* __restrict__ Binv,
                         float* __restrict__ eh, int nnz) {
  int wid = blockIdx.x * (blockDim.x >> 5) + (threadIdx.x >> 5);
  if (wid >= nnz) return;
  int lane = threadIdx.x & 31;
  int nd = node[wid], he = hedge[wid];
  int head = lane >> 3;
  float an = alpha[wid * 4 + head] / (s[he * 4 + head] + 1e-16f);  // normalized alpha
  float coeff = Binv[he] * an;
  if ((lane & 7) == 0) alpha[wid * 4 + head] = an;  // cache normalized alpha for pass 2
  int c = lane * 8;
  const float* xp = xh + (size_t)nd * 256 + c;
  float* ep = eh + (size_t)he * 256 + c;
#pragma unroll
  for (int i = 0; i < 8; ++i) atomicAdd(&ep[i], coeff * xp[i]);
}

// ---------------- layer 1 propagate: hyperedges -> nodes ----------------
__global__ void prop2_l1(const int* __restrict__ node, const int* __restrict__ hedge,
                         const float* __restrict__ eh, const float* __restrict__ alpha,
                         const float* __restrict__ Dinv, float* __restrict__ out1, int nnz) {
  int wid = blockIdx.x * (blockDim.x >> 5) + (threadIdx.x >> 5);
  if (wid >= nnz) return;
  int lane = threadIdx.x & 31;
  int nd = node[wid], he = hedge[wid];
  int head = lane >> 3;
  float coeff = Dinv[nd] * alpha[wid * 4 + head];
  int c = lane * 8;
  const float* ep = eh + (size_t)he * 256 + c;
  float* op = out1 + (size_t)nd * 256 + c;
#pragma unroll
  for (int i = 0; i < 8; ++i) atomicAdd(&op[i], coeff * ep[i]);
}

__global__ void elu_bias(float* __restrict__ h, const float* __restrict__ b, long n, int C) {
  long i = (long)blockIdx.x * blockDim.x + threadIdx.x;
  long st = (long)gridDim.x * blockDim.x;
  for (; i < n; i += st) {
    float v = h[i] + b[i % C];
    h[i] = v > 0.f ? v : (__expf(v) - 1.f);
  }
}

// ---------------- layer 2 propagates (64 ch, no attention) ----------------
__global__ void prop1_l2(const int* __restrict__ node, const int* __restrict__ hedge,
                         const float* __restrict__ xh2, const float* __restrict__ Binv,
                         float* __restrict__ eh2, int nnz) {
  int wid = blockIdx.x * (blockDim.x >> 5) + (threadIdx.x >> 5);
  if (wid >= nnz) return;
  int lane = threadIdx.x & 31;
  int nd = node[wid], he = hedge[wid];
  float coeff = Binv[he];
  int c = lane * 2;
  const float* xp = xh2 + (size_t)nd * 64 + c;
  float* ep = eh2 + (size_t)he * 64 + c;
  atomicAdd(&ep[0], coeff * xp[0]);
  atomicAdd(&ep[1], coeff * xp[1]);
}

__global__ void prop2_l2(const int* __restrict__ node, const int* __restrict__ hedge,
                         const float* __restrict__ eh2, const float* __restrict__ Dinv,
                         float* __restrict__ out, int nnz) {
  int wid = blockIdx.x * (blockDim.x >> 5) + (threadIdx.x >> 5);
  if (wid >= nnz) return;
  int lane = threadIdx.x & 31;
  int nd = node[wid], he = hedge[wid];
  float coeff = Dinv[nd];
  int c = lane * 2;
  const float* ep = eh2 + (size_t)he * 64 + c;
  float* op = out + (size_t)nd * 64 + c;
  atomicAdd(&op[0], coeff * ep[0]);
  atomicAdd(&op[1], coeff * ep[1]);
}

__global__ void bias_add(float* __restrict__ p, const float* __restrict__ b, long n, int C) {
  long i = (long)blockIdx.x * blockDim.x + threadIdx.x;
  long st = (long)gridDim.x * blockDim.x;
  for (; i < n; i += st) p[i] += b[i % C];
}

// ---------------- driver ----------------
extern "C" void kernel_launch(void* const* d_in, const int* in_sizes, int n_in,
                              void* d_out, int out_size, void* d_ws, size_t ws_size,
                              hipStream_t stream) {
  const int CIN = 128, HF = 256, COUT = 64;
  const int N   = in_sizes[0] / CIN;
  const int NNZ = in_sizes[1] / 2;
  const int E   = in_sizes[2] / CIN;

  const float* x    = (const float*)d_in[0];
  const int*   node = (const int*)d_in[1];
  const int*   hedg = node + NNZ;
  const float* hea  = (const float*)d_in[2];
  const float* W1   = (const float*)d_in[3];
  const float* att  = (const float*)d_in[4];
  const float* b1   = (const float*)d_in[5];
  const float* W2   = (const float*)d_in[6];
  const float* b2   = (const float*)d_in[7];
  float* out = (float*)d_out;
  float* ws  = (float*)d_ws;

  // workspace layout (floats); total ~24.5M floats (~98 MB)
  size_t o = 0;
  float* xh  = ws + o; o += (size_t)N * HF;     // xh -> reused as out1 -> h
  float* he  = ws + o; o += (size_t)E * HF;
  float* an  = ws + o; o += (size_t)N * 4;
  float* ae  = ws + o; o += (size_t)E * 4;
  float* alp = ws + o; o += (size_t)NNZ * 4;    // logits -> exp -> normalized alpha
  float* xh2 = ws + o; o += (size_t)N * COUT;
  size_t zoff = o;                              // contiguous zero-init block below
  float*    Dd  = ws + o; o += (size_t)N;
  float*    Bd  = ws + o; o += (size_t)E;
  unsigned* mu  = (unsigned*)(ws + o); o += (size_t)E * 4;
  float*    sb  = ws + o; o += (size_t)E * 4;
  float*    eh  = ws + o; o += (size_t)E * HF;
  float*    eh2 = ws + o; o += (size_t)E * COUT;
  const long zlen = (long)(o - zoff);

  auto cdiv = [](long a, long b) { return (int)((a + b - 1) / b); };

  // zero accumulators (D,B,max,sum,eh,eh2 contiguous) and the output
  fill_f32<<<4096, 256, 0, stream>>>(ws + zoff, zlen, 0.f);
  fill_f32<<<1024, 256, 0, stream>>>(out, (long)N * COUT, 0.f);

  // dense linears on the WMMA pipe (fp32-exact, 16x16x4); M is 16-divisible here
  wmma_gemm<128, 256><<<N / 16, 256, 0, stream>>>(x,   W1, xh, N);
  wmma_gemm<128, 256><<<E / 16, 256, 0, stream>>>(hea, W1, he, E);

  // attention logits pieces
  att_dot<<<(N + 7) / 8, 256, 0, stream>>>(xh, att, an, N, 0);
  att_dot<<<(E + 7) / 8, 256, 0, stream>>>(he, att, ae, E, 64);

  // degrees and segment softmax over hyperedges
  degree_count<<<cdiv(NNZ, 256), 256, 0, stream>>>(node, hedg, Dd, Bd, NNZ);
  logit_max<<<cdiv(NNZ, 256), 256, 0, stream>>>(node, hedg, an, ae, alp, mu, NNZ);
  exp_sum<<<cdiv(NNZ, 256), 256, 0, stream>>>(hedg, alp, mu, sb, NNZ);
  inv_pos<<<256, 256, 0, stream>>>(Dd, (long)(N + E));   // Dd,Bd contiguous

  // layer 1 propagates (L2-resident gather/scatter)
  prop1_l1<<<cdiv(NNZ, 8), 256, 0, stream>>>(node, hedg, xh, alp, sb, Bd, eh, NNZ);
  fill_f32<<<4096, 256, 0, stream>>>(xh, (long)N * HF, 0.f);   // xh becomes out1
  prop2_l1<<<cdiv(NNZ, 8), 256, 0, stream>>>(node, hedg, eh, alp, Dd, xh, NNZ);
  elu_bias<<<4096, 256, 0, stream>>>(xh, b1, (long)N * HF, HF);

  // layer 2: linear + unweighted propagates
  wmma_gemm<256, 64><<<N / 16, 128, 0, stream>>>(xh, W2, xh2, N);
  prop1_l2<<<cdiv(NNZ, 8), 256, 0, stream>>>(node, hedg, xh2, Bd, eh2, NNZ);
  prop2_l2<<<cdiv(NNZ, 8), 256, 0, stream>>>(node, hedg, eh2, Dd, out, NNZ);
  bias_add<<<1024, 256, 0, stream>>>(out, b2, (long)N * COUT, COUT);
}